// GATv2_88776974008615
// MI455X (gfx1250) — compile-verified
//
#include <hip/hip_runtime.h>

#define N_NODES 50000
#define N_EDGES 800000
#define IN_DIM 256
#define EDGE_DIM 64
#define EMBED 128
#define HEADS 8
#define HEAD_DIM 16

typedef __attribute__((ext_vector_type(2))) float    v2f;
typedef __attribute__((ext_vector_type(8))) float    v8f;
typedef __attribute__((ext_vector_type(4))) unsigned u32x4;
typedef __attribute__((ext_vector_type(8))) int      i32x8;
typedef __attribute__((ext_vector_type(4))) int      i32x4;

__device__ __forceinline__ float mish_f(float x) {
    // x * tanh(softplus(x)); softplus ~ x for large x
    float sp = (x > 20.f) ? x : log1pf(__expf(x));
    return x * tanhf(sp);
}

// Monotonic float <-> unsigned mapping so atomicMax(unsigned) == float max.
__device__ __forceinline__ unsigned ord_f32(float f) {
    unsigned u = __float_as_uint(f);
    return (u & 0x80000000u) ? ~u : (u | 0x80000000u);
}
__device__ __forceinline__ float unord_f32(unsigned e) {
    unsigned u = (e & 0x80000000u) ? (e ^ 0x80000000u) : ~e;
    return __uint_as_float(u);
}

__device__ __forceinline__ void atomic_add_f32(float* p, float v) {
    __hip_atomic_fetch_add(p, v, __ATOMIC_RELAXED, __HIP_MEMORY_SCOPE_AGENT);
}

// ---------------------------------------------------------------------------
// K0: zero output accumulators + segment-softmax state
// ---------------------------------------------------------------------------
__global__ void gat_init_kernel(float* __restrict__ out,
                                float* __restrict__ denom,
                                unsigned* __restrict__ segmax) {
    int i = blockIdx.x * blockDim.x + threadIdx.x;
    if (i < N_NODES * EMBED) out[i] = 0.f;
    if (i < N_NODES * HEADS) {
        denom[i] = 0.f;
        segmax[i] = 0u;   // ord encoding: 0 < ord(x) for all x, acts as -inf
    }
}

// ---------------------------------------------------------------------------
// K1: h = X @ W + b   ([50000,256]x[256,128]) via V_WMMA_F32_16X16X4_F32
// One wave per 16x16 output tile; 8 waves / block.
// A frag (32-bit, 16x4): lanes 0-15: M=lane, {K=k0,k0+1}; lanes 16-31: {k0+2,k0+3}
// B frag (4x16, col-major striping): lane&15 = N, same K split as A
// C/D: vgpr v -> M = v + 8*(lane>>4), N = lane&15
// ---------------------------------------------------------------------------
__global__ void node_gemm_kernel(const float* __restrict__ X,
                                 const float* __restrict__ W,
                                 const float* __restrict__ bias,
                                 float* __restrict__ h) {
    const int lane  = threadIdx.x & 31;
    const int wave  = threadIdx.x >> 5;
    const int wgid  = blockIdx.x * 8 + wave;      // 25000 waves total
    const int mtile = wgid >> 3;                  // 0..3124
    const int ntile = wgid & 7;                   // 0..7
    const int lrow  = lane & 15;
    const int khalf = (lane >> 4) << 1;           // 0 or 2
    const int m = mtile * 16 + lrow;
    const int n = ntile * 16 + lrow;

    v8f acc = {};
    for (int k0 = 0; k0 < IN_DIM; k0 += 4) {
        const int kb = k0 + khalf;                // even -> float2 aligned
        const float2 av = *(const float2*)(X + m * IN_DIM + kb);
        v2f a; a.x = av.x; a.y = av.y;
        v2f b; b.x = W[kb * EMBED + n];
               b.y = W[(kb + 1) * EMBED + n];
        acc = __builtin_amdgcn_wmma_f32_16x16x4_f32(
            false, a, false, b, (short)0, acc, false, false);
    }

    const float bv = bias[n];
    const int rbase = mtile * 16 + 8 * (lane >> 4);
    #pragma unroll
    for (int v = 0; v < 8; ++v)
        h[(rbase + v) * EMBED + n] = acc[v] + bv;
}

// ---------------------------------------------------------------------------
// K2 (fused): per 16-edge tile, e_proj = EF@We via WMMA (We staged in LDS
// via the Tensor Data Mover), add gathered h[send]+h[recv]+bias, mish,
// per-head dot with 'a' (N-tile == head since HEAD_DIM==16), cross-lane
// reduce -> logits, ordered atomicMax for segment max.
// Avoids the 410MB e_proj intermediate entirely.
// ---------------------------------------------------------------------------
__global__ void edge_logits_kernel(const float* __restrict__ EF,
                                   const float* __restrict__ We,
                                   const float* __restrict__ We_bias,
                                   const float* __restrict__ attn_a,
                                   const float* __restrict__ h,
                                   const int* __restrict__ senders,
                                   const int* __restrict__ receivers,
                                   float* __restrict__ logits,
                                   unsigned* __restrict__ segmax) {
    __shared__ float lWe[EDGE_DIM * EMBED];   // 32 KB, reused by every tile
    __shared__ float lbias[EMBED];
    __shared__ float la[EMBED];

#if defined(__gfx1250__) && __has_builtin(__builtin_amdgcn_tensor_load_to_lds)
    // TDM: one DMA moves the whole 64x128 f32 We tile into LDS (6-arg form).
    if (threadIdx.x < 32) {
        const unsigned lds_off = (unsigned)(size_t)(&lWe[0]); // flat addr low 32b = LDS offset
        const unsigned long long ga = (unsigned long long)(size_t)We;
        u32x4 g0;
        g0.x = 1u;                                        // count=1 (valid), user mode
        g0.y = lds_off;                                   // lds_addr
        g0.z = (unsigned)(ga & 0xFFFFFFFFu);              // global_addr[31:0]
        g0.w = (unsigned)((ga >> 32) & 0x01FFFFFFu)       // global_addr[56:32]
             | (2u << 30);                                // type=2 ("image")
        i32x8 g1;
        g1[0] = (int)(2u << 16);      // workgroup_mask=0, data_size=2 (4B)
        g1[1] = (int)(128u << 16);    // tensor_dim0[15:0]=128   (bits 63:48)
        g1[2] = (int)(64u << 16);     // tensor_dim0 hi=0, tensor_dim1[15:0]=64
        g1[3] = (int)(128u << 16);    // tensor_dim1 hi=0, tile_dim0=128
        g1[4] = 64;                   // tile_dim1=64, tile_dim2=0
        g1[5] = 128;                  // tensor_dim0_stride[31:0]=128
        g1[6] = 0;                    // stride hi / dim1_stride (unused, 2D)
        g1[7] = 0;
        i32x4 gz4 = {0, 0, 0, 0};     // groups 2/3 unused (<=2D tensor)
        i32x8 gz8 = {0, 0, 0, 0, 0, 0, 0, 0};
        __builtin_amdgcn_tensor_load_to_lds(g0, g1, gz4, gz4, gz8, 0);
        __builtin_amdgcn_s_wait_tensorcnt((unsigned short)0);
    }
#else
    for (int i = threadIdx.x; i < EDGE_DIM * EMBED; i += blockDim.x)
        lWe[i] = We[i];
#endif
    if (threadIdx.x < EMBED) {
        lbias[threadIdx.x] = We_bias[threadIdx.x];
        la[threadIdx.x]    = attn_a[threadIdx.x];
    }
    __syncthreads();

    const int lane   = threadIdx.x & 31;
    const int wave   = threadIdx.x >> 5;
    const int etile  = blockIdx.x * 8 + wave;     // 50000 tiles total
    const int e0     = etile * 16;
    const int lrow   = lane & 15;
    const int laneHi = lane >> 4;
    const int khalf  = laneHi << 1;

    v8f acc[HEADS];
    #pragma unroll
    for (int t = 0; t < HEADS; ++t) { v8f z = {}; acc[t] = z; }

    const int m = e0 + lrow;
    for (int k0 = 0; k0 < EDGE_DIM; k0 += 4) {
        const int kb = k0 + khalf;
        const float2 av = *(const float2*)(EF + m * EDGE_DIM + kb);
        v2f a; a.x = av.x; a.y = av.y;
        #pragma unroll
        for (int t = 0; t < HEADS; ++t) {
            const int n = t * 16 + lrow;
            v2f b; b.x = lWe[kb * EMBED + n];
                   b.y = lWe[(kb + 1) * EMBED + n];
            acc[t] = __builtin_amdgcn_wmma_f32_16x16x4_f32(
                false, a, false, b, (short)0, acc[t], false, false);
        }
    }

    // Epilogue: gather + mish + per-head dot + lane-group reduce
    #pragma unroll
    for (int v = 0; v < 8; ++v) {
        const int edge = e0 + v + 8 * laneHi;     // row M of the C tile
        const int s = senders[edge];
        const int r = receivers[edge];
        #pragma unroll
        for (int t = 0; t < HEADS; ++t) {
            const int col = t * 16 + lrow;        // lanes 0..15 coalesced
            float x = acc[t][v] + lbias[col]
                    + h[s * EMBED + col] + h[r * EMBED + col];
            x = mish_f(x);
            float p = x * la[col];                // a[head t][dim lrow]
            p += __shfl_xor(p, 8, 32);            // reduce within 16-lane half
            p += __shfl_xor(p, 4, 32);
            p += __shfl_xor(p, 2, 32);
            p += __shfl_xor(p, 1, 32);
            if (lrow == 0) {
                logits[edge * HEADS + t] = p;
                atomicMax(&segmax[r * HEADS + t], ord_f32(p));
            }
        }
    }
}

// ---------------------------------------------------------------------------
// K3: ex = exp(logit - segmax[recv]); denom[recv] += ex (in place over logits)
// ---------------------------------------------------------------------------
__global__ void softmax_norm_kernel(const int* __restrict__ receivers,
                                    const unsigned* __restrict__ segmax,
                                    float* __restrict__ logits,
                                    float* __restrict__ denom) {
    int i = blockIdx.x * blockDim.x + threadIdx.x;
    if (i >= N_EDGES * HEADS) return;
    const int e = i >> 3;
    const int t = i & 7;
    const int r = receivers[e];
    const float mx = unord_f32(segmax[r * HEADS + t]);
    const float ex = __expf(logits[i] - mx);
    logits[i] = ex;
    atomic_add_f32(&denom[r * HEADS + t], ex);
}

// ---------------------------------------------------------------------------
// K4: out[recv] += (ex/denom[recv]) * h[send]
// One wave per edge; each lane owns a float4 chunk (32 lanes x 4 = 128 feats).
// b128 gather of h rows + native global_atomic_add_f32 scatter.
// ---------------------------------------------------------------------------
__global__ void scatter_kernel(const int* __restrict__ senders,
                               const int* __restrict__ receivers,
                               const float* __restrict__ ex,
                               const float* __restrict__ denom,
                               const float* __restrict__ h,
                               float* __restrict__ out) {
    const int gid = blockIdx.x * blockDim.x + threadIdx.x;   // 25.6M threads
    const int e = gid >> 5;
    if (e >= N_EDGES) return;
    const int q = (gid & 31) << 2;          // feature base: 0,4,...,124
    const int t = q >> 4;                   // head (4-aligned chunk stays in one head)
    const int r = receivers[e];
    const int s = senders[e];
    const float w = ex[e * HEADS + t] / denom[r * HEADS + t];
    const float4 hv = *(const float4*)(h + s * EMBED + q);
    float* o = out + r * EMBED + q;
    atomic_add_f32(o + 0, w * hv.x);
    atomic_add_f32(o + 1, w * hv.y);
    atomic_add_f32(o + 2, w * hv.z);
    atomic_add_f32(o + 3, w * hv.w);
}

// ---------------------------------------------------------------------------
extern "C" void kernel_launch(void* const* d_in, const int* in_sizes, int n_in,
                              void* d_out, int out_size, void* d_ws, size_t ws_size,
                              hipStream_t stream) {
    const float* X   = (const float*)d_in[0];   // node_features [N,256]
    const float* EF  = (const float*)d_in[1];   // edge_features [E,64]
    const float* W   = (const float*)d_in[2];   // W_kernel [256,128]
    const float* Wb  = (const float*)d_in[3];   // W_bias [128]
    const float* We  = (const float*)d_in[4];   // We_kernel [64,128]
    const float* Web = (const float*)d_in[5];   // We_bias [128]
    const float* av  = (const float*)d_in[6];   // a [8,16]
    const int*   snd = (const int*)d_in[7];     // senders [E]
    const int*   rcv = (const int*)d_in[8];     // receivers [E]
    float* out = (float*)d_out;                 // [N,128]

    char* ws = (char*)d_ws;                     // ~54.4 MB total
    float*    h      = (float*)ws;    ws += sizeof(float)    * (size_t)N_NODES * EMBED;
    float*    logits = (float*)ws;    ws += sizeof(float)    * (size_t)N_EDGES * HEADS;
    unsigned* segmax = (unsigned*)ws; ws += sizeof(unsigned) * (size_t)N_NODES * HEADS;
    float*    denom  = (float*)ws;

    gat_init_kernel<<<(N_NODES * EMBED + 255) / 256, 256, 0, stream>>>(out, denom, segmax);
    node_gemm_kernel<<<3125, 256, 0, stream>>>(X, W, Wb, h);               // 25000 waves
    edge_logits_kernel<<<6250, 256, 0, stream>>>(EF, We, Web, av, h, snd, rcv,
                                                 logits, segmax);          // 50000 tiles
    softmax_norm_kernel<<<(N_EDGES * HEADS + 255) / 256, 256, 0, stream>>>(rcv, segmax,
                                                                           logits, denom);
    const int nscat = (int)(((long long)N_EDGES * 32 + 255) / 256);        // wave per edge
    scatter_kernel<<<nscat, 256, 0, stream>>>(snd, rcv, logits, denom, h, out);
}